// BaseValidation_9174050144277
// MI455X (gfx1250) — compile-verified
//
#include <hip/hip_runtime.h>
#include <cstdint>

// Problem constants from the reference (H,W,P fixed by the model; B derived at launch).
namespace {
constexpr int Hc = 480;
constexpr int Wc = 640;
constexpr int Nc = Hc * Wc;        // 307200, multiple of 256 -> blocks never straddle a batch
constexpr int Pc = 4;
constexpr float EPSc = 1e-9f;
}

// ---------------------------------------------------------------------------
// Native non-returning FP32 global atomic add at device scope (L2 RMW).
// Guarantees global_atomic_add_f32 (no CAS-loop fallback).
// ---------------------------------------------------------------------------
__device__ __forceinline__ void gatomic_fadd(float* p, float v) {
    asm volatile("global_atomic_add_f32 %0, %1, off scope:SCOPE_DEV"
                 :: "v"(p), "v"(v)
                 : "memory");
}

// ---------------------------------------------------------------------------
// Zero-fill (float4 stores).
// ---------------------------------------------------------------------------
__global__ __launch_bounds__(256) void warp_zero_f4(float4* __restrict__ p, int n4) {
    int idx = blockIdx.x * blockDim.x + threadIdx.x;
    if (idx < n4) p[idx] = make_float4(0.f, 0.f, 0.f, 0.f);
}

// ---------------------------------------------------------------------------
// Splat kernel: one thread per (b, pixel). Flow staged via CDNA5 async
// global->LDS loads (ASYNCcnt), splat via device-scope f32 atomics into L2.
// sumx = d_out[0..BN), sumy = d_out[BN..2BN), wsum = d_ws.
// ---------------------------------------------------------------------------
__global__ __launch_bounds__(256) void warp_splat(
    const float* __restrict__ fmx,   // [B,P,H,W]
    const float* __restrict__ fmy,   // [B,P,H,W]
    const int*   __restrict__ pi,    // scalar i
    const int*   __restrict__ ptref, // scalar tref
    float* __restrict__ sumx,        // [B,N]
    float* __restrict__ sumy,        // [B,N]
    float* __restrict__ wsum)        // [B,N]
{
    __shared__ float s_fx[256];
    __shared__ float s_fy[256];

    const int tid = threadIdx.x;
    const int gid = blockIdx.x * 256 + tid;
    const int b   = gid / Nc;          // uniform within a block (Nc % 256 == 0)
    const int p   = gid - b * Nc;

    const int   ii = pi[0];
    const float dt = (float)(ptref[0] - ii);

    const size_t srcoff = ((size_t)(b * Pc + ii)) * (size_t)Nc + (size_t)p;

    // --- CDNA5 async global->LDS staging (tracked by ASYNCcnt) -------------
    {
        const float* ax = fmx + srcoff;
        const float* ay = fmy + srcoff;
        unsigned ldx = (unsigned)(uintptr_t)&s_fx[tid];  // low 32 bits = LDS byte offset
        unsigned ldy = (unsigned)(uintptr_t)&s_fy[tid];
        asm volatile("global_load_async_to_lds_b32 %0, %1, off"
                     :: "v"(ldx), "v"(ax) : "memory");
        asm volatile("global_load_async_to_lds_b32 %0, %1, off"
                     :: "v"(ldy), "v"(ay) : "memory");
    }

    // Overlap index math with the async transfer.
    const int y = p / Wc;
    const int x = p - y * Wc;

    asm volatile("s_wait_asynccnt 0x0" ::: "memory");
    const float fx = s_fx[tid];   // each lane reads back its own slot
    const float fy = s_fy[tid];

    const float wy = (float)y + dt * fy;
    const float wx = (float)x + dt * fx;

    // purge_unfeasible: out-of-image warps contribute exactly zero -> skip.
    if (!(wy >= 0.f && wy <= (float)(Hc - 1) && wx >= 0.f && wx <= (float)(Wc - 1)))
        return;

    const float tyf = floorf(wy);
    const float lxf = floorf(wx);
    const float ay = wy - tyf;           // in [0,1)
    const float axw = wx - lxf;          // in [0,1)
    const int iy = (int)tyf;             // >= 0 (wy >= 0)
    const int ix = (int)lxf;             // >= 0

    const float w00 = (1.f - ay) * (1.f - axw);
    const float w01 = (1.f - ay) * axw;
    const float w10 = ay * (1.f - axw);
    const float w11 = ay * axw;

    // Flattened indices; replicate JAX mode='drop' on [0, N) of the FLAT index
    // (including the x==W row-wrap aliasing the reference exhibits).
    const int f00 = iy * Wc + ix;
    const int f01 = f00 + 1;
    const int f10 = f00 + Wc;
    const int f11 = f10 + 1;

    float* __restrict__ sx = sumx + (size_t)b * Nc;
    float* __restrict__ sy = sumy + (size_t)b * Nc;
    float* __restrict__ sw = wsum + (size_t)b * Nc;

    if (f00 < Nc) {
        gatomic_fadd(&sw[f00], w00);
        gatomic_fadd(&sy[f00], w00 * fy);
        gatomic_fadd(&sx[f00], w00 * fx);
    }
    if (f01 < Nc) {
        gatomic_fadd(&sw[f01], w01);
        gatomic_fadd(&sy[f01], w01 * fy);
        gatomic_fadd(&sx[f01], w01 * fx);
    }
    if (f10 < Nc) {
        gatomic_fadd(&sw[f10], w10);
        gatomic_fadd(&sy[f10], w10 * fy);
        gatomic_fadd(&sx[f10], w10 * fx);
    }
    if (f11 < Nc) {
        gatomic_fadd(&sw[f11], w11);
        gatomic_fadd(&sy[f11], w11 * fy);
        gatomic_fadd(&sx[f11], w11 * fx);
    }
}

// ---------------------------------------------------------------------------
// Finalize: out[k] = out[k] / (wsum[k % BN] + eps), in place over 2*BN elems.
// ---------------------------------------------------------------------------
__global__ __launch_bounds__(256) void warp_finalize(
    float* __restrict__ out, const float* __restrict__ wsum, int bn)
{
    int idx = blockIdx.x * blockDim.x + threadIdx.x;
    if (idx < 2 * bn) {
        int k = (idx < bn) ? idx : (idx - bn);
        out[idx] = out[idx] / (wsum[k] + EPSc);
    }
}

// ---------------------------------------------------------------------------
// Launcher
// ---------------------------------------------------------------------------
extern "C" void kernel_launch(void* const* d_in, const int* in_sizes, int n_in,
                              void* d_out, int out_size, void* d_ws, size_t ws_size,
                              hipStream_t stream) {
    const float* fmx   = (const float*)d_in[0];   // [B,P,H,W]
    const float* fmy   = (const float*)d_in[1];   // [B,P,H,W]
    const int*   pi    = (const int*)d_in[2];     // scalar i
    const int*   ptref = (const int*)d_in[3];     // scalar tref

    const int B  = in_sizes[0] / (Pc * Nc);       // derive batch from input size
    const int bn = B * Nc;                        // per-output element count

    float* out  = (float*)d_out;                  // [2*bn]: x first, then y
    float* sumx = out;                            // accumulate w*fx here
    float* sumy = out + (size_t)bn;               // accumulate w*fy here
    float* wsum = (float*)d_ws;                   // [bn] weight accumulator

    // 1) zero accumulators (harness poisons buffers; we re-init every call)
    {
        const int n4a = (2 * bn) / 4;
        const int n4b = bn / 4;
        warp_zero_f4<<<(n4a + 255) / 256, 256, 0, stream>>>((float4*)out, n4a);
        warp_zero_f4<<<(n4b + 255) / 256, 256, 0, stream>>>((float4*)wsum, n4b);
    }

    // 2) bilinear forward splat (bn threads, Nc % 256 == 0 so blocks stay in-batch)
    warp_splat<<<bn / 256, 256, 0, stream>>>(fmx, fmy, pi, ptref, sumx, sumy, wsum);

    // 3) normalize in place
    warp_finalize<<<(2 * bn + 255) / 256, 256, 0, stream>>>(out, wsum, bn);
}